// ALIFNetwork_24610162606570
// MI455X (gfx1250) — compile-verified
//
#include <hip/hip_runtime.h>

// ---------------------------------------------------------------------------
// ALIF e-prop network, MI455X (gfx1250, wave32).
// Design notes:
//  * e_v / e_v_in are rank-1 (column-constant) -> stored as vectors.
//  * Matvecs done with V_WMMA_F32_16X16X4_F32 (fp32 exact path; memory bound,
//    so WMMA lane waste is free and it keeps VALU available).
//  * Everything (67MB state + 22.5MB weights) is L2-resident on the 192MB L2.
//  * 4 kernels per timestep, 500 steps, all on `stream`; no atomicAdd(float)
//    so results are bitwise deterministic across replays.
//  * k-loop is split per weight matrix and 2-way unrolled with statically
//    assigned accumulators so no EXEC-mask branches surround the inner WMMAs.
// ---------------------------------------------------------------------------

typedef float v2f __attribute__((ext_vector_type(2)));
typedef float v8f __attribute__((ext_vector_type(8)));

constexpr int   N_IN  = 700;
constexpr int   N_REC = 2048;
constexpr int   N_OUT = 20;
constexpr int   T_STEPS = 500;
constexpr float ALPHA  = 0.99501247919268232f;   // exp(-1/200)
constexpr float K_OUTC = 0.95122942450071402f;   // exp(-1/20)
constexpr float THR    = 0.6f;
constexpr float DT     = 1.0f;
constexpr float N_REF  = 5.0f;
constexpr float BETA   = 0.07f;
constexpr float GAMMA  = 0.3f;
constexpr float LR     = 0.01f;

// ---- workspace layout (float offsets) ----
constexpr size_t WS_V      = 0;
constexpr size_t WS_A      = 2048;
constexpr size_t WS_Z      = 4096;
constexpr size_t WS_TSLP   = 6144;
constexpr size_t WS_PD     = 8192;
constexpr size_t WS_L      = 10240;
constexpr size_t WS_EV     = 12288;
constexpr size_t WS_EVIN   = 14336;                 // 700 used
constexpr size_t WS_Y      = 15360;                 // 20 used
constexpr size_t WS_ONEHOT = 15424;                 // 20 used
constexpr size_t WS_EA     = 16384;                                    // 2048x2048
constexpr size_t WS_FTR    = WS_EA   + (size_t)N_REC * N_REC;          // 2048x2048
constexpr size_t WS_EAIN   = WS_FTR  + (size_t)N_REC * N_REC;          // 700x2048
constexpr size_t WS_FTRIN  = WS_EAIN + (size_t)N_IN  * N_REC;          // 700x2048
constexpr size_t WS_TROUT  = WS_FTRIN+ (size_t)N_IN  * N_REC;          // 2048x20
constexpr size_t WS_TOTAL  = WS_TROUT+ (size_t)N_REC * N_OUT;          // 11,313,152 f

// ---- output layout (float offsets, reference return order) ----
constexpr size_t OUT_OUTS  = 0;                                        // 500x20
constexpr size_t OUT_VOLT  = OUT_OUTS  + (size_t)T_STEPS * N_OUT;      // 500x2048
constexpr size_t OUT_SPK   = OUT_VOLT  + (size_t)T_STEPS * N_REC;
constexpr size_t OUT_ADAPT = OUT_SPK   + (size_t)T_STEPS * N_REC;
constexpr size_t OUT_GWIN  = OUT_ADAPT + (size_t)T_STEPS * N_REC;      // 700x2048
constexpr size_t OUT_GWREC = OUT_GWIN  + (size_t)N_IN  * N_REC;        // 2048x2048
constexpr size_t OUT_GWOUT = OUT_GWREC + (size_t)N_REC * N_REC;        // 2048x20
constexpr size_t OUT_GB    = OUT_GWOUT + (size_t)N_REC * N_OUT;        // 20
constexpr size_t GRAD_N    = OUT_GB + N_OUT - OUT_GWIN;

// ---------------------------------------------------------------------------
// init: zero workspace + gradient outputs, set tslp = N_REF. Runs every call.
// ---------------------------------------------------------------------------
__global__ __launch_bounds__(256) void k_init(float* __restrict__ ws,
                                              float* __restrict__ out) {
  size_t id = (size_t)blockIdx.x * blockDim.x + threadIdx.x;
  size_t stride = (size_t)gridDim.x * blockDim.x;
  for (size_t k = id; k < WS_TOTAL; k += stride) {
    float v = 0.0f;
    if (k >= WS_TSLP && k < WS_TSLP + N_REC) v = N_REF;
    ws[k] = v;
  }
  for (size_t k = id; k < GRAD_N; k += stride) out[OUT_GWIN + k] = 0.0f;
}

// ---------------------------------------------------------------------------
// One WMMA GEMV step: accumulate s[r..r+3] (dot) W[r..r+3, jb..jb+15].
// A layout (16x4 f32): lane = m + 16*(k>=2), vgpr = k&1 -> replicate the
// vector to all 16 rows (only D row 0 is consumed; other rows are isolated).
// B layout (4x16 f32): vgpr = k&1, lane = n + 16*(k>=2).
// ---------------------------------------------------------------------------
__device__ __forceinline__ void wmma_step(const float* __restrict__ s,
                                          const float* __restrict__ W,
                                          int r, int half, int n, int jb,
                                          v8f& acc) {
  const int rr = r + 2 * half;
  v2f A; A.x = s[rr]; A.y = s[rr + 1];
  const float* wp = W + (size_t)rr * N_REC + jb + n;
  v2f B; B.x = wp[0]; B.y = wp[N_REC];
  acc = __builtin_amdgcn_wmma_f32_16x16x4_f32(false, A, false, B,
                                              (short)0, acc, false, false);
}

// ---------------------------------------------------------------------------
// K_A: r = x_t@w_in + z@w_rec via WMMA f32 16x16x4; then v,a updates.
// Block = 256 threads (8 waves), each block owns 16 output columns.
// Waves split the concatenated K space (175 + 512 k-blocks of 4); each wave's
// range is pre-split into its w_in and w_rec sub-intervals so the inner loops
// are branch-free with static accumulator assignment.
// ---------------------------------------------------------------------------
__global__ __launch_bounds__(256)
void k_matvec(const float* __restrict__ x, const float* __restrict__ w_in,
              const float* __restrict__ w_rec, float* __restrict__ ws,
              float* __restrict__ out, int t) {
  const int tid  = threadIdx.x;
  const int lane = tid & 31;
  const int wv   = tid >> 5;
  const int jb   = blockIdx.x * 16;
  const float* xt = x + (size_t)t * N_IN;
  const float* z  = ws + WS_Z;

  constexpr int KB_IN  = N_IN / 4;            // 175
  constexpr int KB_TOT = KB_IN + N_REC / 4;   // 687
  constexpr int PER    = (KB_TOT + 7) / 8;    // 86 k-blocks per wave
  const int kb0 = wv * PER;
  const int kb1 = (kb0 + PER < KB_TOT) ? kb0 + PER : KB_TOT;

  const int half = lane >> 4;    // selects K pair {0,1} vs {2,3}
  const int n    = lane & 15;    // output column within tile

  // intersect this wave's range with the w_in part [0,175) / w_rec part
  const int a0 = (kb0 < KB_IN) ? kb0 : KB_IN;
  const int a1 = (kb1 < KB_IN) ? kb1 : KB_IN;
  const int b0 = ((kb0 > KB_IN) ? kb0 : KB_IN) - KB_IN;
  const int b1 = ((kb1 > KB_IN) ? kb1 : KB_IN) - KB_IN;

  v8f acc0 = {};
  v8f acc1 = {};
  int kb = a0;
  for (; kb + 2 <= a1; kb += 2) {
    wmma_step(xt, w_in, kb * 4,     half, n, jb, acc0);
    wmma_step(xt, w_in, kb * 4 + 4, half, n, jb, acc1);
  }
  if (kb < a1) wmma_step(xt, w_in, kb * 4, half, n, jb, acc0);
  kb = b0;
  for (; kb + 2 <= b1; kb += 2) {
    wmma_step(z, w_rec, kb * 4,     half, n, jb, acc0);
    wmma_step(z, w_rec, kb * 4 + 4, half, n, jb, acc1);
  }
  if (kb < b1) wmma_step(z, w_rec, kb * 4, half, n, jb, acc0);

  v8f acc = acc0 + acc1;

  __shared__ float red[8][16];
  if (lane < 16) red[wv][lane] = acc[0];   // D[0, n] lives in c[0], lanes 0-15
  __syncthreads();
  if (tid < 16) {
    float r = 0.0f;
#pragma unroll
    for (int w2 = 0; w2 < 8; ++w2) r += red[w2][tid];
    const int j = jb + tid;
    const float zj = z[j];
    const float vj = ALPHA * ws[WS_V + j] + r - THR * zj;  // old z
    const float aj = ALPHA * ws[WS_A + j] + zj;
    ws[WS_V + j] = vj;
    ws[WS_A + j] = aj;
    out[OUT_VOLT  + (size_t)t * N_REC + j] = vj;
    out[OUT_ADAPT + (size_t)t * N_REC + j] = aj;
  }
}

// ---------------------------------------------------------------------------
// K_B: single workgroup. Global refractory reduction, spike branch, pd, ev,
// readout+softmax, err, gb_out, tr_out/gw_out, learning signal L = err @ B.
// ---------------------------------------------------------------------------
__global__ __launch_bounds__(1024)
void k_spike(const float* __restrict__ x, const int* __restrict__ target,
             const float* __restrict__ w_out, const float* __restrict__ b_out,
             const float* __restrict__ Bfb, const int* __restrict__ training,
             float* __restrict__ ws, float* __restrict__ out, int t) {
  const int tid = threadIdx.x;
  const int train = training[0];
  __shared__ int   s_flag;
  __shared__ float s_y[N_OUT];
  __shared__ float s_err[N_OUT];
  if (tid == 0) s_flag = 0;
  __syncthreads();

  // any(tslp_old < N_REF) over all 2048 neurons
  bool anyr = false;
  for (int j = tid; j < N_REC; j += 1024) anyr |= (ws[WS_TSLP + j] < N_REF);
  if (anyr) atomicOr(&s_flag, 1);
  __syncthreads();
  const int refr = s_flag;

  for (int j = tid; j < N_REC; j += 1024) {
    const float vj = ws[WS_V + j];
    const float aj = ws[WS_A + j];
    float ts = ws[WS_TSLP + j];
    float zj;
    if (refr) {
      ts = (ts < N_REF) ? ts + DT : ts;
      zj = (ts > N_REF) ? 1.0f : 0.0f;
    } else {
      zj = (vj >= THR + BETA * aj) ? 1.0f : 0.0f;
      if (zj == 1.0f) ts = 0.0f;
    }
    ws[WS_Z + j] = zj;
    ws[WS_TSLP + j] = ts;
    out[OUT_SPK + (size_t)t * N_REC + j] = zj;
    if (train) {
      const float pdj = (GAMMA / THR) *
          fmaxf(0.0f, 1.0f - fabsf((vj - THR - BETA * aj) / THR));
      ws[WS_PD + j] = pdj;
      ws[WS_EV + j] = ALPHA * ws[WS_EV + j] + zj;   // rank-1 e_v
    }
  }
  if (train && tid < N_IN)
    ws[WS_EVIN + tid] = ALPHA * ws[WS_EVIN + tid] + x[(size_t)t * N_IN + tid];
  __syncthreads();

  // readout: y = K_OUT*y + z@w_out + b_out  (one wave per output)
  if (tid < N_OUT * 32) {
    const int o = tid >> 5, ln = tid & 31;
    float p = 0.0f;
    for (int j = ln; j < N_REC; j += 32)
      p += ws[WS_Z + j] * w_out[(size_t)j * N_OUT + o];
#pragma unroll
    for (int off = 16; off; off >>= 1) p += __shfl_xor(p, off, 32);
    if (ln == 0) {
      const float yn = K_OUTC * ws[WS_Y + o] + p + b_out[o];
      ws[WS_Y + o] = yn;
      s_y[o] = yn;
    }
  }
  __syncthreads();

  if (tid == 0) {
    if (train) ws[WS_ONEHOT + target[t]] = 1.0f;   // cumulative one-hot
    float m = s_y[0];
    for (int o = 1; o < N_OUT; ++o) m = fmaxf(m, s_y[o]);
    float e[N_OUT];
    float ssum = 0.0f;
    for (int o = 0; o < N_OUT; ++o) { e[o] = expf(s_y[o] - m); ssum += e[o]; }
    for (int o = 0; o < N_OUT; ++o) {
      const float so = e[o] / ssum;
      out[OUT_OUTS + (size_t)t * N_OUT + o] = so;
      const float er = so - ws[WS_ONEHOT + o];
      s_err[o] = er;
      if (train) out[OUT_GB + o] += LR * er;
    }
  }
  __syncthreads();

  if (train) {
    for (int j = tid; j < N_REC; j += 1024) {
      const float zj = ws[WS_Z + j];
#pragma unroll
      for (int o = 0; o < N_OUT; ++o) {
        const size_t idx = (size_t)j * N_OUT + o;
        const float tr = ALPHA * ws[WS_TROUT + idx] + zj;
        ws[WS_TROUT + idx] = tr;
        out[OUT_GWOUT + idx] += LR * s_err[o] * tr;
      }
      float Ls = 0.0f;
#pragma unroll
      for (int o = 0; o < N_OUT; ++o) Ls += s_err[o] * Bfb[(size_t)o * N_REC + j];
      ws[WS_L + j] = Ls;
    }
  }
}

// ---------------------------------------------------------------------------
// K_C/K_D: fused e_a / ftr / gw recurrence (bandwidth bound, float4 over j).
//   ea'  = pd[j]*ev[i] + (ALPHA - pd[j]*BETA)*ea
//   tr   = pd[j]*(ev[i] - BETA*ea')
//   ftr' = ALPHA*ftr + tr
//   gw  += LR*L[j]*ftr'
// ---------------------------------------------------------------------------
__global__ __launch_bounds__(256)
void k_traces(const float* __restrict__ evv, float* __restrict__ ea,
              float* __restrict__ ftr, float* __restrict__ gw,
              const float* __restrict__ ws, const int* __restrict__ training,
              int rows) {
  if (!training[0]) return;
  const int gid = blockIdx.x * 256 + threadIdx.x;
  const int i  = gid >> 9;           // 512 float4 per 2048-wide row
  const int j4 = (gid & 511) << 2;
  if (i >= rows) return;
  const float evi = evv[i];
  const float4 pd4 = *(const float4*)(ws + WS_PD + j4);
  const float4 L4  = *(const float4*)(ws + WS_L + j4);
  const size_t idx = (size_t)i * N_REC + j4;
  float4 ea4 = *(float4*)(ea + idx);
  float4 ft4 = *(float4*)(ftr + idx);
  float4 gw4 = *(float4*)(gw + idx);

#define TRACE1(C)                                            \
  {                                                          \
    const float pd = pd4.C;                                  \
    float e = pd * evi + (ALPHA - pd * BETA) * ea4.C;        \
    float tr = pd * (evi - BETA * e);                        \
    float f = ALPHA * ft4.C + tr;                            \
    ea4.C = e; ft4.C = f;                                    \
    gw4.C = gw4.C + LR * L4.C * f;                           \
  }
  TRACE1(x) TRACE1(y) TRACE1(z) TRACE1(w)
#undef TRACE1

  *(float4*)(ea + idx) = ea4;
  *(float4*)(ftr + idx) = ft4;
  *(float4*)(gw + idx) = gw4;
}

// ---------------------------------------------------------------------------
extern "C" void kernel_launch(void* const* d_in, const int* in_sizes, int n_in,
                              void* d_out, int out_size, void* d_ws, size_t ws_size,
                              hipStream_t stream) {
  const float* x      = (const float*)d_in[0];
  const int*   target = (const int*)  d_in[1];
  const float* w_in   = (const float*)d_in[2];
  const float* w_rec  = (const float*)d_in[3];
  const float* w_out  = (const float*)d_in[4];
  const float* b_out  = (const float*)d_in[5];
  const float* Bfb    = (const float*)d_in[6];
  const int*   train  = (const int*)  d_in[7];
  float* out = (float*)d_out;
  float* ws  = (float*)d_ws;

  k_init<<<1024, 256, 0, stream>>>(ws, out);

  for (int t = 0; t < T_STEPS; ++t) {
    k_matvec<<<N_REC / 16, 256, 0, stream>>>(x, w_in, w_rec, ws, out, t);
    k_spike<<<1, 1024, 0, stream>>>(x, target, w_out, b_out, Bfb, train, ws, out, t);
    k_traces<<<(N_REC * (N_REC / 4)) / 256, 256, 0, stream>>>(
        ws + WS_EV, ws + WS_EA, ws + WS_FTR, out + OUT_GWREC, ws, train, N_REC);
    k_traces<<<(N_IN * (N_REC / 4)) / 256, 256, 0, stream>>>(
        ws + WS_EVIN, ws + WS_EAIN, ws + WS_FTRIN, out + OUT_GWIN, ws, train, N_IN);
  }
}